// LaplacianLoss_27608049779003
// MI455X (gfx1250) — compile-verified
//
#include <hip/hip_runtime.h>
#include <hip/hip_bf16.h>

// LaplacianLoss on MI455X (gfx1250).
// Memory-bound 7-point-stencil + squared-error reduction.
// CDNA5 path: Tensor Data Mover (tensor_load_to_lds) stages 8KB z-plane slabs
// into an LDS ring buffer asynchronously (TENSORcnt), overlapped with compute.

typedef unsigned int u32;
typedef u32 u32x4 __attribute__((ext_vector_type(4)));
typedef u32 u32x8 __attribute__((ext_vector_type(8)));

#define NN 128
#define TY 14          // interior y rows per tile (9 tiles cover y=1..126)
#define CZ 14          // interior z planes per chunk (9 chunks cover z=1..126)
#define SLAB_ROWS (TY + 2)

// Issue one TDM 2-D tile load: 16 rows x 128 f32 (row stride 128 elements)
// from global `gptr` into LDS byte-offset `lds_off`. Wave-level op, EXEC ignored.
__device__ __forceinline__ void tdm_load_plane(const float* gptr, u32 lds_off) {
    unsigned long long ga = (unsigned long long)(size_t)gptr;
    // D# group 0: [1:0]=count=1, [63:32]=lds_addr, [120:64]=global_addr, [127:126]=type=2
    u32x4 g0 = { 1u,
                 lds_off,
                 (u32)ga,
                 (u32)((ga >> 32) & 0x01FFFFFFu) | (2u << 30) };
    // D# group 1:
    //  dw0: workgroup_mask=0, data_size([17:16])=2 -> 4 bytes
    //  dw1: barrier_addr=0 | tensor_dim0.lo16<<16  (tensor_dim0 = 128)
    //  dw2: tensor_dim0.hi16 | tensor_dim1.lo16<<16 (tensor_dim1 = 16)
    //  dw3: tensor_dim1.hi16 | tile_dim0<<16        (tile_dim0 = 128)
    //  dw4: tile_dim1 | tile_dim2<<16               (tile_dim1 = 16, tile_dim2 = 0)
    //  dw5: tensor_dim0_stride.lo32                 (stride = 128 elements)
    //  dw6: stride0.hi16 | stride1.lo16<<16 = 0
    //  dw7: stride1.hi32 = 0
    u32x8 g1 = { 2u << 16,
                 128u << 16,
                 16u << 16,
                 128u << 16,
                 16u,
                 128u,
                 0u,
                 0u };
    asm volatile("tensor_load_to_lds %0, %1" :: "s"(g0), "s"(g1) : "memory");
}

__global__ __launch_bounds__(256) void lap_mse_kernel(
        const float* __restrict__ field,     // "output" tensor (B,1,128,128,128)
        const float* __restrict__ data,      // "data" tensor
        float* __restrict__ partial) {
    __shared__ float slab[4][SLAB_ROWS][NN];   // 4-slot ring, 8KB per plane slab
    __shared__ float red[256];

    const int tid = threadIdx.x;
    const int zc  = blockIdx.x;     // z chunk  (0..8)
    const int yt  = blockIdx.y;     // y tile   (0..8)
    const int b   = blockIdx.z;     // batch    (0..15)
    const int z0  = zc * CZ;        // plane p maps to global z = z0 + p
    const int y0  = yt * TY;        // slab row 0 maps to global y = y0

    const float* f = field + (size_t)b * NN * NN * NN;
    const bool wave0 = (tid >> 5) == 0;

    // Prime the pipeline: planes 0,1,2 -> slots 0,1,2
    if (wave0) {
        for (int p = 0; p < 3; ++p) {
            tdm_load_plane(f + ((size_t)(z0 + p) * NN + y0) * NN,
                           (u32)(size_t)&slab[p][0][0]);
        }
    }

    float acc = 0.0f;
    for (int p = 1; p <= CZ; ++p) {
        // All waves done reading slot (p+2)&3 (== (p-2)&3, last read at iter p-1)
        __syncthreads();
        if (wave0) {
            if (p + 2 <= CZ + 1) {
                tdm_load_plane(f + ((size_t)(z0 + p + 2) * NN + y0) * NN,
                               (u32)(size_t)&slab[(p + 2) & 3][0][0]);
                __builtin_amdgcn_s_wait_tensorcnt(1);  // planes <= p+1 complete
            } else {
                __builtin_amdgcn_s_wait_tensorcnt(0);  // drain tail
            }
        }
        __syncthreads();   // publish TDM-written LDS to all waves

        const int s0 = (p - 1) & 3, s1 = p & 3, s2 = (p + 1) & 3;
        const float* drow = data + ((size_t)b * NN + (size_t)(z0 + p)) * NN * NN;

        for (int idx = tid; idx < TY * 126; idx += 256) {
            const int r = idx / 126;            // 0..13 -> slab row r+1
            const int x = idx - r * 126 + 1;    // 1..126
            const float c  = slab[s1][r + 1][x];
            const float s6 = slab[s1][r + 1][x - 1] + slab[s1][r + 1][x + 1]
                           + slab[s1][r][x]         + slab[s1][r + 2][x]
                           + slab[s0][r + 1][x]     + slab[s2][r + 1][x];
            const float lap  = (s6 - 6.0f * c) * 16384.0f;   // 1/dx^2 = 128^2
            const float dval = __builtin_nontemporal_load(
                                   &drow[(size_t)(y0 + r + 1) * NN + x]);
            const float diff = lap + dval;
            acc += diff * diff;
        }
    }

    // Block reduction (deterministic tree)
    red[tid] = acc;
    __syncthreads();
    for (int off = 128; off > 0; off >>= 1) {
        if (tid < off) red[tid] += red[tid + off];
        __syncthreads();
    }
    if (tid == 0)
        partial[((size_t)b * gridDim.y + yt) * gridDim.x + zc] = red[0];
}

__global__ __launch_bounds__(256) void finalize_kernel(
        const float* __restrict__ partial, int n, float* __restrict__ out) {
    __shared__ double red[256];
    double a = 0.0;
    for (int i = threadIdx.x; i < n; i += 256) a += (double)partial[i];
    red[threadIdx.x] = a;
    __syncthreads();
    for (int off = 128; off > 0; off >>= 1) {
        if (threadIdx.x < off) red[threadIdx.x] += red[threadIdx.x + off];
        __syncthreads();
    }
    if (threadIdx.x == 0) {
        // mean over 16*126^3 points, * WEIGHT(0.2); LX^2*LY^2*LZ^2 = 1
        out[0] = (float)(red[0] * (0.2 / 32006016.0));
    }
}

extern "C" void kernel_launch(void* const* d_in, const int* in_sizes, int n_in,
                              void* d_out, int out_size, void* d_ws, size_t ws_size,
                              hipStream_t stream) {
    (void)in_sizes; (void)n_in; (void)out_size; (void)ws_size;
    const float* field = (const float*)d_in[0];   // 'output' in setup_inputs order
    const float* data  = (const float*)d_in[1];   // 'data'
    float* partial = (float*)d_ws;                // 1296 floats of scratch

    dim3 grid(9, 9, 16);   // z-chunks, y-tiles, batch -> 1296 blocks
    lap_mse_kernel<<<grid, 256, 0, stream>>>(field, data, partial);
    finalize_kernel<<<1, 256, 0, stream>>>(partial, 9 * 9 * 16, (float*)d_out);
}